// Resampling_8615704396582
// MI455X (gfx1250) — compile-verified
//
#include <hip/hip_runtime.h>
#include <hip/hip_bf16.h>

typedef __attribute__((ext_vector_type(2))) float v2f;
typedef __attribute__((ext_vector_type(4))) float v4f;
typedef __attribute__((ext_vector_type(8))) float v8f;

// Problem constants
#define NB 16
#define NP 8
#define NH 32
#define NW 32
#define ND 32
#define NC 8
#define NPTS (NH * NW * ND)          // 32768 points per (b,p)
#define VOXELS_PER_BP (NPTS * NC)    // 262144 floats per (b,p) volume

__global__ __launch_bounds__(256) void
Resampling_8615704396582_kernel(const float* __restrict__ fmap,
                                const float* __restrict__ theta,
                                float* __restrict__ out) {
    const int tid  = threadIdx.x;
    const int lane = tid & 31;
    const int bp   = blockIdx.x >> 7;                       // 128 blocks per (b,p)
    const int pbase = ((blockIdx.x & 127) << 8) + (tid & ~31); // wave's first point id

    // ---- theta (block-uniform -> scalar loads) ----
    const float* th = theta + bp * 12;
    const float t00 = th[0],  t01 = th[1],  t02 = th[2],  t03 = th[3];
    const float t10 = th[4],  t11 = th[5],  t12 = th[6],  t13 = th[7];
    const float t20 = th[8],  t21 = th[9],  t22 = th[10], t23 = th[11];

    // ---- Build WMMA A operand: theta padded to 16x4 ----
    // 32-bit A 16x4 layout: lane holds row M = lane&15; VGPR0 = K0(lo)/K2(hi), VGPR1 = K1(lo)/K3(hi)
    const int  M  = lane & 15;
    const bool hi = lane >= 16;
    float a0 = 0.0f, a1 = 0.0f;
    if (M == 0)      { a0 = hi ? t02 : t00; a1 = hi ? t03 : t01; }
    else if (M == 1) { a0 = hi ? t12 : t10; a1 = hi ? t13 : t11; }
    else if (M == 2) { a0 = hi ? t22 : t20; a1 = hi ? t23 : t21; }
    v2f A = {a0, a1};

    // ---- Build WMMA B operands: grid columns (h,w,d,1) for 2 groups of 16 points ----
    // B 4x16 layout (mirror of A): lane holds col N = lane&15; VGPR0 = K0(lo)/K2(hi), VGPR1 = K1(lo)/K3(hi)
    const int n = lane & 15;
    v2f B0, B1;
    {
        int q0 = pbase + n;            // group 0 point
        int q1 = pbase + 16 + n;       // group 1 point
        float h0 = (float)(q0 >> 10), w0 = (float)((q0 >> 5) & 31), d0 = (float)(q0 & 31);
        float h1 = (float)(q1 >> 10), w1 = (float)((q1 >> 5) & 31), d1 = (float)(q1 & 31);
        B0[0] = hi ? d0 : h0;          // K=2 : K=0
        B0[1] = hi ? 1.0f : w0;        // K=3 : K=1
        B1[0] = hi ? d1 : h1;
        B1[1] = hi ? 1.0f : w1;
    }

    // ---- Coordinate transform on the matrix unit: D = theta_pad * grid_tile ----
    v8f Z = {};
    v8f C0 = __builtin_amdgcn_wmma_f32_16x16x4_f32(false, A, false, B0, (short)0, Z, false, false);
    v8f C1 = __builtin_amdgcn_wmma_f32_16x16x4_f32(false, A, false, B1, (short)0, Z, false, false);

    // D row M lives in VGPR M (lanes 0-15) -> y/x/z of point n are D[0..2] at lane n.
    // Redistribute so each lane owns its own point's coords.
    float y_s, x_s, z_s;
    {
        float ylo = __shfl(C0[0], n, 32), yhi = __shfl(C1[0], n, 32);
        float xlo = __shfl(C0[1], n, 32), xhi = __shfl(C1[1], n, 32);
        float zlo = __shfl(C0[2], n, 32), zhi = __shfl(C1[2], n, 32);
        y_s = hi ? yhi : ylo;
        x_s = hi ? xhi : xlo;
        z_s = hi ? zhi : zlo;
    }

    const int idx = pbase + lane;      // this thread's point within (b,p)

    // ---- floors + out-of-bounds handling (matches reference) ----
    int x0 = (int)floorf(x_s);
    int y0 = (int)floorf(y_s);
    int z0 = (int)floorf(z_s);
    const bool outside = (x0 < 0) | (x0 >= 31) | (y0 < 0) | (y0 >= 31) | (z0 < 0) | (z0 >= 31);
    if (outside) { x0 = 0; y0 = 0; z0 = 0; }
    const int x1 = x0 + 1, y1 = y0 + 1;

    const float xd = x_s - (float)x0;
    const float yd = y_s - (float)y0;
    const float zd = z_s - (float)z0;
    const float wx0 = 1.0f - xd, wy0 = 1.0f - yd, wz0 = 1.0f - zd;

    const float w000 = wx0 * wy0 * wz0;
    const float w100 = xd  * wy0 * wz0;
    const float w010 = wx0 * yd  * wz0;
    const float w001 = wx0 * wy0 * zd;
    const float w101 = xd  * wy0 * zd;
    const float w011 = wx0 * yd  * zd;
    const float w110 = xd  * yd  * wz0;
    const float w111 = xd  * yd  * zd;

    // ---- gather 8 corners (z0/z1 pairs are 64B contiguous) ----
    const float* src = fmap + (size_t)bp * VOXELS_PER_BP;
    const float* p00 = src + (size_t)(((y0 * 32 + x0) * 32 + z0) * 8); // c000,c001
    const float* p01 = src + (size_t)(((y0 * 32 + x1) * 32 + z0) * 8); // c100,c101
    const float* p10 = src + (size_t)(((y1 * 32 + x0) * 32 + z0) * 8); // c010,c011
    const float* p11 = src + (size_t)(((y1 * 32 + x1) * 32 + z0) * 8); // c110,c111

    v4f c000a = *(const v4f*)(p00),      c000b = *(const v4f*)(p00 + 4);
    v4f c001a = *(const v4f*)(p00 + 8),  c001b = *(const v4f*)(p00 + 12);
    v4f c100a = *(const v4f*)(p01),      c100b = *(const v4f*)(p01 + 4);
    v4f c101a = *(const v4f*)(p01 + 8),  c101b = *(const v4f*)(p01 + 12);
    v4f c010a = *(const v4f*)(p10),      c010b = *(const v4f*)(p10 + 4);
    v4f c011a = *(const v4f*)(p10 + 8),  c011b = *(const v4f*)(p10 + 12);
    v4f c110a = *(const v4f*)(p11),      c110b = *(const v4f*)(p11 + 4);
    v4f c111a = *(const v4f*)(p11 + 8),  c111b = *(const v4f*)(p11 + 12);

    // ---- trilinear blend (reference summation order) ----
    v4f accA = c000a * w000;
    v4f accB = c000b * w000;
    accA += c100a * w100;  accB += c100b * w100;
    accA += c010a * w010;  accB += c010b * w010;
    accA += c001a * w001;  accB += c001b * w001;
    accA += c101a * w101;  accB += c101b * w101;
    accA += c011a * w011;  accB += c011b * w011;
    accA += c110a * w110;  accB += c110b * w110;
    accA += c111a * w111;  accB += c111b * w111;

    // ---- streamed output: non-temporal so it doesn't evict the input from L2 ----
    float* dst = out + ((size_t)bp * NPTS + (size_t)idx) * NC;
    __builtin_nontemporal_store(accA, (v4f*)dst);
    __builtin_nontemporal_store(accB, (v4f*)(dst + 4));
}

extern "C" void kernel_launch(void* const* d_in, const int* in_sizes, int n_in,
                              void* d_out, int out_size, void* d_ws, size_t ws_size,
                              hipStream_t stream) {
    const float* fmap  = (const float*)d_in[0];  // (16,8,32,32,32,8) fp32
    const float* theta = (const float*)d_in[1];  // (16,8,3,4) fp32
    float* out = (float*)d_out;                  // (16,8,32,32,32,8) fp32

    // 128 (b,p) pairs * 128 blocks each = 16384 blocks of 256 threads (1 thread/point)
    dim3 grid(NB * NP * (NPTS / 256));
    dim3 block(256);
    Resampling_8615704396582_kernel<<<grid, block, 0, stream>>>(fmap, theta, out);
}